// DecoderxLSTMBlock_45921790328952
// MI455X (gfx1250) — compile-verified
//
#include <hip/hip_runtime.h>
#include <math.h>

typedef __attribute__((ext_vector_type(2))) float v2f;
typedef __attribute__((ext_vector_type(4))) float v4f;
typedef __attribute__((ext_vector_type(8))) float v8f;

// ---------------------------------------------------------------------------
// WMMA GEMM:  Y[m,n] = act( (sum_k X[m,k]*W[n,k] + bias[n]) * scale )
// X: (M x K) row-major, W: (N x K) row-major (i.e. Y = X @ W^T).
// One wave32 per 16x16 output tile; V_WMMA_F32_16X16X4_F32 over K in steps of 4.
// A layout (16x4 f32): lanes0-15 = M rows, VGPR0/1 = K{0,1}; lanes16-31 = K{2,3}.
// B layout (4x16 f32): symmetric: lanes0-15 hold K{0,1} of column n, lanes16-31 K{2,3}.
// ---------------------------------------------------------------------------
template <int ACT>   // 0 = identity, 1 = tanh
__global__ void wmma_gemm(const float* __restrict__ X, const float* __restrict__ W,
                          const float* __restrict__ bias, float* __restrict__ Y,
                          int K, int N, float scale) {
    const int ntn  = N >> 4;
    const int tm   = blockIdx.x / ntn;
    const int tn   = blockIdx.x % ntn;
    const int lane = threadIdx.x;          // 0..31, wave32
    const int l    = lane & 15;
    const int koff = (lane >> 4) << 1;     // 0 or 2

    const float* __restrict__ xp = X + (size_t)(tm * 16 + l) * K + koff;
    const float* __restrict__ wp = W + (size_t)(tn * 16 + l) * K + koff;

    v8f acc = {};
#pragma unroll 4
    for (int k = 0; k < K; k += 4) {
        v2f a = *(const v2f*)(xp + k);
        v2f b = *(const v2f*)(wp + k);
        acc = __builtin_amdgcn_wmma_f32_16x16x4_f32(false, a, false, b,
                                                    (short)0, acc, false, false);
    }

    // C layout: VGPR r -> row M = r (lanes0-15) / r+8 (lanes16-31), col N = lane&15
    const int   col   = tn * 16 + l;
    const float bv    = bias[col];
    const int   rbase = tm * 16 + ((lane >> 4) << 3);
#pragma unroll
    for (int r = 0; r < 8; ++r) {
        float t = (acc[r] + bv) * scale;
        if (ACT == 1) t = tanhf(t);
        Y[(size_t)(rbase + r) * N + col] = t;
    }
}

// Dual-input GEMM: Y = X1@W1^T + X2@W2^T + bias  (no activation)
__global__ void wmma_gemm2(const float* __restrict__ X1, const float* __restrict__ W1,
                           const float* __restrict__ X2, const float* __restrict__ W2,
                           const float* __restrict__ bias, float* __restrict__ Y,
                           int K, int N) {
    const int ntn  = N >> 4;
    const int tm   = blockIdx.x / ntn;
    const int tn   = blockIdx.x % ntn;
    const int lane = threadIdx.x;
    const int l    = lane & 15;
    const int koff = (lane >> 4) << 1;

    const float* __restrict__ x1 = X1 + (size_t)(tm * 16 + l) * K + koff;
    const float* __restrict__ w1 = W1 + (size_t)(tn * 16 + l) * K + koff;
    const float* __restrict__ x2 = X2 + (size_t)(tm * 16 + l) * K + koff;
    const float* __restrict__ w2 = W2 + (size_t)(tn * 16 + l) * K + koff;

    v8f acc = {};
#pragma unroll 4
    for (int k = 0; k < K; k += 4) {
        v2f a = *(const v2f*)(x1 + k);
        v2f b = *(const v2f*)(w1 + k);
        acc = __builtin_amdgcn_wmma_f32_16x16x4_f32(false, a, false, b,
                                                    (short)0, acc, false, false);
    }
#pragma unroll 4
    for (int k = 0; k < K; k += 4) {
        v2f a = *(const v2f*)(x2 + k);
        v2f b = *(const v2f*)(w2 + k);
        acc = __builtin_amdgcn_wmma_f32_16x16x4_f32(false, a, false, b,
                                                    (short)0, acc, false, false);
    }

    const int   col   = tn * 16 + l;
    const float bv    = bias[col];
    const int   rbase = tm * 16 + ((lane >> 4) << 3);
#pragma unroll
    for (int r = 0; r < 8; ++r)
        Y[(size_t)(rbase + r) * N + col] = acc[r] + bv;
}

// ---------------------------------------------------------------------------
// Block reduction helper (blockDim.x == 256)
// ---------------------------------------------------------------------------
__device__ __forceinline__ float block_reduce_sum(float v, float* sbuf) {
    const int tid = threadIdx.x;
    sbuf[tid] = v;
    __syncthreads();
#pragma unroll
    for (int s = 128; s > 0; s >>= 1) {
        if (tid < s) sbuf[tid] += sbuf[tid + s];
        __syncthreads();
    }
    float r = sbuf[0];
    __syncthreads();
    return r;
}

__device__ __forceinline__ float sigmoidf_(float x) {
    return 1.0f / (1.0f + __expf(-x));
}

// LayerNorm over rows of length 1024; one 256-thread block per row.
__global__ void ln_kernel(const float* __restrict__ X, const float* __restrict__ g,
                          const float* __restrict__ bt, float* __restrict__ Y, int Hn) {
    __shared__ float sbuf[256];
    const int row = blockIdx.x;
    const float* __restrict__ x = X + (size_t)row * Hn;
    float* __restrict__ y = Y + (size_t)row * Hn;
    float xs[4];
    float s = 0.f;
#pragma unroll
    for (int u = 0; u < 4; ++u) { xs[u] = x[threadIdx.x + u * 256]; s += xs[u]; }
    s = block_reduce_sum(s, sbuf);
    const float mu = s / (float)Hn;
    float vs = 0.f;
#pragma unroll
    for (int u = 0; u < 4; ++u) { float d = xs[u] - mu; vs += d * d; }
    vs = block_reduce_sum(vs, sbuf);
    const float inv = rsqrtf(vs / (float)Hn + 1e-5f);
#pragma unroll
    for (int u = 0; u < 4; ++u) {
        int idx = threadIdx.x + u * 256;
        y[idx] = (xs[u] - mu) * inv * g[idx] + bt[idx];
    }
}

// sLSTM pointwise: gates from pre (B,4H), states out, h_res = x + s_h_new
__global__ void slstm_kernel(const float* __restrict__ pre, const float* __restrict__ x,
                             const float* __restrict__ s_c0,
                             float* __restrict__ s_h_new, float* __restrict__ s_c_new,
                             float* __restrict__ s_n_new, float* __restrict__ h_res, int Hn) {
    const int idx = blockIdx.x * blockDim.x + threadIdx.x;   // b*Hn + h
    const int b = idx / Hn, h = idx - b * Hn;
    const float* __restrict__ p = pre + (size_t)b * 4 * Hn;
    const float z = tanhf(p[h]);
    const float i = __expf(p[Hn + h]);
    const float f = sigmoidf_(p[2 * Hn + h]);
    const float o = sigmoidf_(p[3 * Hn + h]);
    const float cn = f * s_c0[idx] + i * z;
    const float nn = f + i;                // s_n init == 1
    const float hn = o * (cn / nn);
    s_c_new[idx] = cn;
    s_n_new[idx] = nn;
    s_h_new[idx] = hn;
    h_res[idx]   = x[idx] + hn;
}

// Scalar i/f gates: mi = exp(m_in.w_mi + b_mi), mf = sigmoid(m_in.w_mf + b_mf)
__global__ void mif_kernel(const float* __restrict__ m_in, const float* __restrict__ w_mi,
                           const float* __restrict__ b_mi, const float* __restrict__ w_mf,
                           const float* __restrict__ b_mf, float* __restrict__ mi_out,
                           float* __restrict__ mf_out, int Hn) {
    __shared__ float sbuf[256];
    const int row = blockIdx.x;
    const float* __restrict__ x = m_in + (size_t)row * Hn;
    float si = 0.f, sf = 0.f;
#pragma unroll
    for (int u = 0; u < 4; ++u) {
        int idx = threadIdx.x + u * 256;
        float xv = x[idx];
        si += xv * w_mi[idx];
        sf += xv * w_mf[idx];
    }
    si = block_reduce_sum(si, sbuf);
    sf = block_reduce_sum(sf, sbuf);
    if (threadIdx.x == 0) {
        mi_out[row] = __expf(si + b_mi[0]);
        mf_out[row] = sigmoidf_(sf + b_mf[0]);
    }
}

// Per-row mLSTM readout using rank-1 structure of m_C_new:
//   num = mf*m_diag*q + mi*v*(k.q);  den = max(|mf*(m_n.q)+mi*(k.q)|,1)
//   h_out = LN_o(h_res + sigmoid(mo_pre)*num/den);  m_n_new = mf*m_n + mi*k
__global__ void mlstm_kernel(const float* __restrict__ q, const float* __restrict__ k,
                             const float* __restrict__ v, const float* __restrict__ mo_pre,
                             const float* __restrict__ m_n0, const float* __restrict__ m_diag,
                             const float* __restrict__ h_res, const float* __restrict__ mi_v,
                             const float* __restrict__ mf_v, const float* __restrict__ g,
                             const float* __restrict__ bt, float* __restrict__ m_n_new,
                             float* __restrict__ h_out, int Hn) {
    __shared__ float sbuf[256];
    const int row = blockIdx.x;
    const size_t base = (size_t)row * Hn;
    const float mi = mi_v[row], mf = mf_v[row];

    float qv[4], kv[4], t[4];
    float kq = 0.f, nq0 = 0.f;
#pragma unroll
    for (int u = 0; u < 4; ++u) {
        int idx = threadIdx.x + u * 256;
        qv[u] = q[base + idx];
        kv[u] = k[base + idx];
        kq  += kv[u] * qv[u];
        nq0 += m_n0[base + idx] * qv[u];
    }
    kq  = block_reduce_sum(kq, sbuf);
    nq0 = block_reduce_sum(nq0, sbuf);
    const float den = fmaxf(fabsf(mf * nq0 + mi * kq), 1.0f);

    float s = 0.f;
#pragma unroll
    for (int u = 0; u < 4; ++u) {
        int idx = threadIdx.x + u * 256;
        float mn  = mf * m_n0[base + idx] + mi * kv[u];
        m_n_new[base + idx] = mn;
        float num = mf * m_diag[base + idx] * qv[u] + mi * v[base + idx] * kq;
        float mo  = sigmoidf_(mo_pre[base + idx]);
        t[u] = h_res[base + idx] + mo * (num / den);
        s += t[u];
    }
    s = block_reduce_sum(s, sbuf);
    const float mu = s / (float)Hn;
    float vs = 0.f;
#pragma unroll
    for (int u = 0; u < 4; ++u) { float d = t[u] - mu; vs += d * d; }
    vs = block_reduce_sum(vs, sbuf);
    const float inv = rsqrtf(vs / (float)Hn + 1e-5f);
#pragma unroll
    for (int u = 0; u < 4; ++u) {
        int idx = threadIdx.x + u * 256;
        h_out[base + idx] = (t[u] - mu) * inv * g[idx] + bt[idx];
    }
}

// m_C_new[b,i,j] = mf[b]*(i==j)*m_diag[b,i] + mi[b]*v[b,i]*k[b,j]
// 268 MB streaming write -> 128-bit non-temporal stores (bypass L2 retention).
// grid.x = B*Hn rows, block 256, one v4f per thread.
__global__ void mC_kernel(const float* __restrict__ v, const float* __restrict__ k,
                          const float* __restrict__ m_diag, const float* __restrict__ mi,
                          const float* __restrict__ mf, float* __restrict__ mC, int Hn) {
    const int row = blockIdx.x;            // b*Hn + i
    const int b   = row / Hn;
    const int i   = row - b * Hn;
    const float a = mi[b] * v[row];
    const int j   = threadIdx.x << 2;
    v4f kk = *(const v4f*)(k + (size_t)b * Hn + j);
    v4f out = a * kk;
    if ((i >> 2) == (int)threadIdx.x)      // diagonal element lands in this v4f
        out[i & 3] += mf[b] * m_diag[row];
    __builtin_nontemporal_store(out, (v4f*)(mC + (size_t)row * Hn + j));
}

// ---------------------------------------------------------------------------
extern "C" void kernel_launch(void* const* d_in, const int* in_sizes, int n_in,
                              void* d_out, int out_size, void* d_ws, size_t ws_size,
                              hipStream_t stream) {
    const float* x_t    = (const float*)d_in[0];
    const float* base   = (const float*)d_in[1];
    const float* W_in   = (const float*)d_in[2];
    const float* b_in   = (const float*)d_in[3];
    const float* ln_s_g = (const float*)d_in[4];
    const float* ln_s_b = (const float*)d_in[5];
    const float* ln_m_g = (const float*)d_in[6];
    const float* ln_m_b = (const float*)d_in[7];
    const float* ln_o_g = (const float*)d_in[8];
    const float* ln_o_b = (const float*)d_in[9];
    const float* W_sx   = (const float*)d_in[10];
    const float* W_sh   = (const float*)d_in[11];
    const float* b_s    = (const float*)d_in[12];
    const float* W_q    = (const float*)d_in[13];
    const float* b_q    = (const float*)d_in[14];
    const float* W_k    = (const float*)d_in[15];
    const float* b_k    = (const float*)d_in[16];
    const float* W_v    = (const float*)d_in[17];
    const float* b_v    = (const float*)d_in[18];
    const float* w_mi   = (const float*)d_in[19];
    const float* b_mi   = (const float*)d_in[20];
    const float* w_mf   = (const float*)d_in[21];
    const float* b_mf   = (const float*)d_in[22];
    const float* W_mo   = (const float*)d_in[23];
    const float* b_mo   = (const float*)d_in[24];
    const float* W_ish  = (const float*)d_in[25];
    const float* b_ish  = (const float*)d_in[26];
    const float* W_isc  = (const float*)d_in[27];
    const float* b_isc  = (const float*)d_in[28];
    const float* W_imd  = (const float*)d_in[29];
    const float* b_imd  = (const float*)d_in[30];
    const float* W_imn  = (const float*)d_in[31];
    const float* b_imn  = (const float*)d_in[32];

    constexpr int Bsz = 64, Din = 512, Hn = 1024;
    constexpr int BH = Bsz * Hn;                      // 65536

    float* ws = (float*)d_ws;
    float* xw     = ws + 0 * BH;
    float* s_h0   = ws + 1 * BH;
    float* s_c0   = ws + 2 * BH;
    float* m_diag = ws + 3 * BH;
    float* m_n0   = ws + 4 * BH;
    float* s_in   = ws + 5 * BH;
    float* h_res  = ws + 6 * BH;
    float* m_in   = ws + 7 * BH;
    float* qb     = ws + 8 * BH;
    float* kb     = ws + 9 * BH;
    float* vb     = ws + 10 * BH;
    float* mo_pre = ws + 11 * BH;
    float* pre    = ws + 12 * BH;                     // (B,4H) = 4*BH
    float* mi_s   = ws + 16 * BH;
    float* mf_s   = ws + 16 * BH + 64;

    float* out      = (float*)d_out;
    float* h_out    = out;
    float* s_h_new  = out + (size_t)BH;
    float* s_c_new  = out + (size_t)2 * BH;
    float* s_n_new  = out + (size_t)3 * BH;
    float* m_C_new  = out + (size_t)4 * BH;
    float* m_n_new  = m_C_new + (size_t)Bsz * Hn * Hn;

    const int tilesH  = (Bsz / 16) * (Hn / 16);       // 256 tiles for N=1024
    const int tiles4H = (Bsz / 16) * (4 * Hn / 16);   // 1024 tiles for N=4096

    // 1) input projection + state initializers (WMMA f32 GEMMs)
    wmma_gemm<0><<<tilesH, 32, 0, stream>>>(x_t,  W_in,  b_in,  xw,     Din, Hn, 1.0f);
    wmma_gemm<1><<<tilesH, 32, 0, stream>>>(base, W_ish, b_ish, s_h0,   Hn,  Hn, 1.0f);
    wmma_gemm<1><<<tilesH, 32, 0, stream>>>(base, W_isc, b_isc, s_c0,   Hn,  Hn, 1.0f);
    wmma_gemm<1><<<tilesH, 32, 0, stream>>>(base, W_imd, b_imd, m_diag, Hn,  Hn, 1.0f);
    wmma_gemm<1><<<tilesH, 32, 0, stream>>>(base, W_imn, b_imn, m_n0,   Hn,  Hn, 1.0f);

    // 2) s_in = LN_s(x)
    ln_kernel<<<Bsz, 256, 0, stream>>>(xw, ln_s_g, ln_s_b, s_in, Hn);

    // 3) pre = s_in@W_sx^T + s_h0@W_sh^T + b_s   (B x 4H)
    wmma_gemm2<<<tiles4H, 32, 0, stream>>>(s_in, W_sx, s_h0, W_sh, b_s, pre, Hn, 4 * Hn);

    // 4) sLSTM gates + states + h_res
    slstm_kernel<<<BH / 256, 256, 0, stream>>>(pre, xw, s_c0, s_h_new, s_c_new,
                                               s_n_new, h_res, Hn);

    // 5) m_in = LN_m(h_res)
    ln_kernel<<<Bsz, 256, 0, stream>>>(h_res, ln_m_g, ln_m_b, m_in, Hn);

    // 6) q/k/v/o projections (k pre-scaled by 1/sqrt(H) = 1/32)
    wmma_gemm<0><<<tilesH, 32, 0, stream>>>(m_in, W_q,  b_q,  qb,     Hn, Hn, 1.0f);
    wmma_gemm<0><<<tilesH, 32, 0, stream>>>(m_in, W_k,  b_k,  kb,     Hn, Hn, 1.0f / 32.0f);
    wmma_gemm<0><<<tilesH, 32, 0, stream>>>(m_in, W_v,  b_v,  vb,     Hn, Hn, 1.0f);
    wmma_gemm<0><<<tilesH, 32, 0, stream>>>(m_in, W_mo, b_mo, mo_pre, Hn, Hn, 1.0f);

    // 7) scalar gates mi/mf
    mif_kernel<<<Bsz, 256, 0, stream>>>(m_in, w_mi, b_mi, w_mf, b_mf, mi_s, mf_s, Hn);

    // 8) mLSTM readout (rank-1 shortcut, no m_C readback) + output LN
    mlstm_kernel<<<Bsz, 256, 0, stream>>>(qb, kb, vb, mo_pre, m_n0, m_diag, h_res,
                                          mi_s, mf_s, ln_o_g, ln_o_b, m_n_new, h_out, Hn);

    // 9) materialize m_C_new (268 MB, non-temporal 128-bit streaming stores)
    mC_kernel<<<Bsz * Hn, 256, 0, stream>>>(vb, kb, m_diag, mi_s, mf_s, m_C_new, Hn);
}